// NodeNetwork_75617194213894
// MI455X (gfx1250) — compile-verified
//
#include <hip/hip_runtime.h>
#include <math.h>

// ---------------------------------------------------------------------------
// NodeNetwork for MI455X (gfx1250, wave32, WMMA).
//   msg = scatter_add(x over edges, both directions)           [N,128]
//   h1  = tanh(LN(msg @ W0[:, :128].T + x @ W0[:,128:].T + b0))  -> d_out
//   h2  = tanh(LN(h1 @ W1.T + b1))                               -> ws (msg)
//   out = h2 @ W2.T + b2                                         -> d_out
// GEMMs run on V_WMMA_F32_16X16X4_F32 (exact fp32 matrix pipe).
// Scatter uses native GLOBAL_ATOMIC_ADD_F32 (unsafeAtomicAdd) when available.
// ---------------------------------------------------------------------------

typedef float v2f __attribute__((ext_vector_type(2)));
typedef float v8f __attribute__((ext_vector_type(8)));

#define LN_EPS 1e-5f
#define DF 128

__device__ __forceinline__ v8f wmma_f32_k4(v2f a, v2f b, v8f c) {
  // 8 args: (neg_a, A, neg_b, B, c_mod, C, reuse_a, reuse_b)
  return __builtin_amdgcn_wmma_f32_16x16x4_f32(false, a, false, b, (short)0, c,
                                               false, false);
}

// SFINAE: use unsafeAtomicAdd (plain HW fp32 global atomic) if declared,
// otherwise fall back to atomicAdd. Detection via dependent-name lookup.
template <typename P>
__device__ __forceinline__ auto atom_add_f32_impl(P p, float v, int)
    -> decltype(unsafeAtomicAdd(p, v), void()) {
  unsafeAtomicAdd(p, v);
}
template <typename P>
__device__ __forceinline__ void atom_add_f32_impl(P p, float v, long) {
  atomicAdd(p, v);
}
__device__ __forceinline__ void atom_add_f32(float* p, float v) {
  atom_add_f32_impl(p, v, 0);
}

__global__ void zero_f32(float* __restrict__ p, long long n) {
  long long i = (long long)blockIdx.x * blockDim.x + threadIdx.x;
  long long stride = (long long)gridDim.x * blockDim.x;
  for (; i < n; i += stride) p[i] = 0.0f;
}

// One wave32 per edge: gather x[start], x[end] (float4/lane), atomic-add the
// reversed direction into msg. 8 f32 atomics per lane, L2-resident working set.
__global__ __launch_bounds__(256) void scatter_edges(
    const float* __restrict__ x, const long long* __restrict__ ei,
    float* __restrict__ msg, int E) {
  int e = blockIdx.x * 8 + (threadIdx.x >> 5);
  if (e >= E) return;
  int lane = threadIdx.x & 31;
  long long s = ei[e];
  long long t = ei[(long long)E + e];
  float4 vs = *(const float4*)(x + s * DF + lane * 4);
  float4 vt = *(const float4*)(x + t * DF + lane * 4);
  float* ms = msg + s * DF + lane * 4;
  float* mt = msg + t * DF + lane * 4;
  atom_add_f32(mt + 0, vs.x);
  atom_add_f32(mt + 1, vs.y);
  atom_add_f32(mt + 2, vs.z);
  atom_add_f32(mt + 3, vs.w);
  atom_add_f32(ms + 0, vt.x);
  atom_add_f32(ms + 1, vt.y);
  atom_add_f32(ms + 2, vt.z);
  atom_add_f32(ms + 3, vt.w);
}

// Block = 8 waves = one 16-row x 128-col output tile. Wave w owns the 16x16
// WMMA tile at columns [16w, 16w+16). A-fragment (16x4 f32): lanes 0-15 carry
// K=k0,k0+1 of row M=lane, lanes 16-31 carry K=k0+2,k0+3. B = W^T loaded as a
// contiguous float2 from W[col][k0+2*kh]. C/D: acc[i] = C[i + 8*kh][col].
template <bool HAS_A1, bool DO_LN>
__global__ __launch_bounds__(256) void gemm16_fused(
    const float* __restrict__ A0, const float* __restrict__ A1,
    const float* __restrict__ W, int ldw, const float* __restrict__ bias,
    const float* __restrict__ gamma, const float* __restrict__ beta,
    float* __restrict__ out, int N) {
  __shared__ __align__(16) float smem[16][DF + 4];

  const int row0 = blockIdx.x * 16;
  const int wave = threadIdx.x >> 5;
  const int lane = threadIdx.x & 31;
  const int lr = lane & 15;
  const int kh = lane >> 4;  // which K-half of the fragment this lane holds
  const int col = wave * 16 + lr;

  int ar = row0 + lr;
  if (ar >= N) ar = N - 1;  // clamp (tile rows divide N exactly anyway)

  const float* arow0 = A0 + (size_t)ar * DF;
  const float* wrow = W + (size_t)col * ldw;

  v8f acc = {};
#pragma unroll
  for (int k0 = 0; k0 < DF; k0 += 4) {
    v2f a = *(const v2f*)(arow0 + k0 + 2 * kh);
    v2f b = *(const v2f*)(wrow + k0 + 2 * kh);
    acc = wmma_f32_k4(a, b, acc);
  }
  if (HAS_A1) {  // concat([msg, x]) @ W0.T == msg@W0[:, :128].T + x@W0[:,128:].T
    const float* arow1 = A1 + (size_t)ar * DF;
    const float* wrow1 = wrow + DF;
#pragma unroll
    for (int k0 = 0; k0 < DF; k0 += 4) {
      v2f a = *(const v2f*)(arow1 + k0 + 2 * kh);
      v2f b = *(const v2f*)(wrow1 + k0 + 2 * kh);
      acc = wmma_f32_k4(a, b, acc);
    }
  }

  const float bv = bias[col];

  if (DO_LN) {
#pragma unroll
    for (int i = 0; i < 8; ++i) smem[i + 8 * kh][col] = acc[i] + bv;
    __syncthreads();

    const float4 g4 = *(const float4*)(gamma + lane * 4);
    const float4 be4 = *(const float4*)(beta + lane * 4);
    // wave w reduces rows 2w and 2w+1 (4 elems/lane, shfl_xor tree, wave32)
#pragma unroll
    for (int rr = 2 * wave; rr < 2 * wave + 2; ++rr) {
      float4 v = *(const float4*)(&smem[rr][lane * 4]);
      float s = v.x + v.y + v.z + v.w;
      float s2 = v.x * v.x + v.y * v.y + v.z * v.z + v.w * v.w;
#pragma unroll
      for (int off = 16; off > 0; off >>= 1) {
        s += __shfl_xor(s, off, 32);
        s2 += __shfl_xor(s2, off, 32);
      }
      float mean = s * (1.0f / DF);
      float var = s2 * (1.0f / DF) - mean * mean;
      float rinv = rsqrtf(var + LN_EPS);
      float4 o;
      o.x = tanhf((v.x - mean) * rinv * g4.x + be4.x);
      o.y = tanhf((v.y - mean) * rinv * g4.y + be4.y);
      o.z = tanhf((v.z - mean) * rinv * g4.z + be4.z);
      o.w = tanhf((v.w - mean) * rinv * g4.w + be4.w);
      int orow = row0 + rr;
      if (orow < N) *(float4*)(out + (size_t)orow * DF + lane * 4) = o;
    }
  } else {
#pragma unroll
    for (int i = 0; i < 8; ++i) {
      int m = row0 + i + 8 * kh;
      if (m < N) out[(size_t)m * DF + col] = acc[i] + bv;
    }
  }
}

extern "C" void kernel_launch(void* const* d_in, const int* in_sizes, int n_in,
                              void* d_out, int out_size, void* d_ws,
                              size_t ws_size, hipStream_t stream) {
  const float* x = (const float*)d_in[0];
  const long long* ei = (const long long*)d_in[1];  // int64 [2, E]
  const float* W0 = (const float*)d_in[2];          // [128, 256]
  const float* b0 = (const float*)d_in[3];
  const float* g0 = (const float*)d_in[4];
  const float* be0 = (const float*)d_in[5];
  const float* W1 = (const float*)d_in[6];  // [128, 128]
  const float* b1 = (const float*)d_in[7];
  const float* g1 = (const float*)d_in[8];
  const float* be1 = (const float*)d_in[9];
  const float* W2 = (const float*)d_in[10];  // [128, 128]
  const float* b2 = (const float*)d_in[11];

  const int N = in_sizes[0] / DF;
  const int E = in_sizes[1] / 2;

  // Buffer ping-pong keeps ws usage to a single [N,128] f32 buffer:
  //   msg (ws) --L1--> h1 (d_out) --L2--> h2 (ws, msg dead) --L3--> d_out
  float* msg = (float*)d_ws;
  float* h1 = (float*)d_out;
  float* h2 = msg;
  float* out = (float*)d_out;

  const long long msg_elems = (long long)N * DF;
  zero_f32<<<4096, 256, 0, stream>>>(msg, msg_elems);
  scatter_edges<<<(E + 7) / 8, 256, 0, stream>>>(x, ei, msg, E);

  const int gblocks = (N + 15) / 16;
  gemm16_fused<true, true>
      <<<gblocks, 256, 0, stream>>>(msg, x, W0, 256, b0, g0, be0, h1, N);
  gemm16_fused<false, true>
      <<<gblocks, 256, 0, stream>>>(h1, nullptr, W1, DF, b1, g1, be1, h2, N);
  gemm16_fused<false, false>
      <<<gblocks, 256, 0, stream>>>(h2, nullptr, W2, DF, b2, nullptr, nullptr,
                                    out, N);
}